// Reacher2D_Graph_Net_17051020165368
// MI455X (gfx1250) — compile-verified
//
#include <hip/hip_runtime.h>
#include <hip/hip_bf16.h>
#include <math.h>

// ---------------------------------------------------------------------------
// CDNA5 (gfx1250) DiffPool GNN forward, fp32 WMMA (V_WMMA_F32_16X16X4_F32).
// One wave (32 lanes) computes one 16x16 output tile, K-loop in steps of 4,
// pointer-bumped addressing, b64 loads for contiguous fragment pairs.
// ---------------------------------------------------------------------------

typedef __attribute__((ext_vector_type(2))) float v2f;
typedef __attribute__((ext_vector_type(8))) float v8f;

__device__ __forceinline__ v8f wmma_f32(v2f a, v2f b, v8f c) {
  // 8 args: (neg_a, A, neg_b, B, c_mod, C, reuse_a, reuse_b)
  return __builtin_amdgcn_wmma_f32_16x16x4_f32(false, a, false, b, (short)0, c,
                                               false, false);
}

// ---------------------------------------------------------------------------
// GEMM: D = op(A) @ B  [+ B[row,:] self-loop] [+ bias] [relu]
// A: TRANSA ? [K x lda] (use A^T) : [M x lda];  B: [K x ldb];  D: [M x ldd]
// M, N multiples of 16; K multiple of 4. Flags are compile-time.
// ---------------------------------------------------------------------------
template <bool TRANSA, bool ADDB, bool HASBIAS, bool RELU>
__global__ __launch_bounds__(256) void k_gemm_t(
    const float* __restrict__ A, int lda,
    const float* __restrict__ B, int ldb,
    const float* __restrict__ bias,
    float* __restrict__ D, int ldd, int M, int N, int K) {
  const int lane = threadIdx.x & 31;
  const int hl = lane >> 4;  // lane half selects K pair {0,1} vs {2,3}
  const int l16 = lane & 15;
  const int tilesN = N >> 4;
  const int tid = blockIdx.x * (blockDim.x >> 5) + (threadIdx.x >> 5);
  if (tid >= (M >> 4) * tilesN) return;
  const int tm = tid / tilesN;
  const int tn = tid - tm * tilesN;
  const int row = (tm << 4) + l16;
  const int col = (tn << 4) + l16;

  // B fragment: column walk, stride ldb per K; two scalar streams.
  const float* bp0 = B + (size_t)(hl << 1) * ldb + col;
  const float* bp1 = bp0 + ldb;
  const size_t bstep = (size_t)ldb << 2;  // 4 rows of B per step

  v8f acc = {};
  if (TRANSA) {
    const float* ap0 = A + (size_t)(hl << 1) * lda + row;
    const float* ap1 = ap0 + lda;
    const size_t astep = (size_t)lda << 2;
#pragma unroll 4
    for (int k0 = 0; k0 < K; k0 += 4) {
      v2f a, b;
      a.x = *ap0;
      a.y = *ap1;
      b.x = *bp0;
      b.y = *bp1;
      acc = wmma_f32(a, b, acc);
      ap0 += astep; ap1 += astep;
      bp0 += bstep; bp1 += bstep;
    }
  } else {
    const float* ap = A + (size_t)row * lda + (hl << 1);  // 8B-aligned pair
#pragma unroll 4
    for (int k0 = 0; k0 < K; k0 += 4) {
      v2f a = *(const v2f*)ap;
      v2f b;
      b.x = *bp0;
      b.y = *bp1;
      acc = wmma_f32(a, b, acc);
      ap += 4;
      bp0 += bstep; bp1 += bstep;
    }
  }
#pragma unroll
  for (int r = 0; r < 8; ++r) {
    const int rr = (tm << 4) + r + (hl << 3);
    float v = acc[r];
    if (ADDB) v += B[(size_t)rr * ldb + col];
    if (HASBIAS) v += bias[col];
    if (RELU) v = fmaxf(v, 0.0f);
    D[(size_t)rr * ldd + col] = v;
  }
}

// ---------------------------------------------------------------------------
// Fused DenseSAGE layer: Out = relu( (Agg * invdeg[row]) @ Wrel + X @ Wroot + b )
// Two WMMAs per K-step. Out written into a column slice (ldout).
// ---------------------------------------------------------------------------
__global__ __launch_bounds__(256) void k_sage(
    const float* __restrict__ Agg, int ldAgg,
    const float* __restrict__ X, int ldX,
    const float* __restrict__ invdeg,
    const float* __restrict__ Wrel, const float* __restrict__ Wroot,
    const float* __restrict__ bias,
    float* __restrict__ Out, int ldout, int M, int N, int K) {
  const int lane = threadIdx.x & 31;
  const int hl = lane >> 4;
  const int l16 = lane & 15;
  const int tilesN = N >> 4;
  const int tid = blockIdx.x * (blockDim.x >> 5) + (threadIdx.x >> 5);
  if (tid >= (M >> 4) * tilesN) return;
  const int tm = tid / tilesN;
  const int tn = tid - tm * tilesN;
  const int row = (tm << 4) + l16;
  const int col = (tn << 4) + l16;
  const float sc = invdeg[row];

  const float* gp = Agg + (size_t)row * ldAgg + (hl << 1);
  const float* xp = X + (size_t)row * ldX + (hl << 1);
  const float* wr0 = Wrel + (size_t)(hl << 1) * N + col;
  const float* wr1 = wr0 + N;
  const float* wo0 = Wroot + (size_t)(hl << 1) * N + col;
  const float* wo1 = wo0 + N;
  const size_t wstep = (size_t)N << 2;

  v8f acc = {};
#pragma unroll 4
  for (int k0 = 0; k0 < K; k0 += 4) {
    v2f a = *(const v2f*)gp;
    a.x *= sc;
    a.y *= sc;
    v2f b;
    b.x = *wr0;
    b.y = *wr1;
    acc = wmma_f32(a, b, acc);
    a = *(const v2f*)xp;
    b.x = *wo0;
    b.y = *wo1;
    acc = wmma_f32(a, b, acc);
    gp += 4; xp += 4;
    wr0 += wstep; wr1 += wstep;
    wo0 += wstep; wo1 += wstep;
  }
#pragma unroll
  for (int r = 0; r < 8; ++r) {
    const int rr = (tm << 4) + r + (hl << 3);
    Out[(size_t)rr * ldout + col] = fmaxf(acc[r] + bias[col], 0.0f);
  }
}

// ---------------------------------------------------------------------------
// Link loss: acc += sum_tile (adj[i,j] - (S @ S^T)[i,j])^2   (WMMA tiles)
// Both operands are contiguous row pairs of S -> b64 loads.
// ---------------------------------------------------------------------------
__global__ __launch_bounds__(256) void k_linkloss(
    const float* __restrict__ S, int lds,
    const float* __restrict__ adj, int M, int Kd,
    float* __restrict__ acc_out) {
  const int lane = threadIdx.x & 31;
  const int hl = lane >> 4;
  const int l16 = lane & 15;
  const int tilesN = M >> 4;
  const int tid = blockIdx.x * (blockDim.x >> 5) + (threadIdx.x >> 5);
  if (tid >= tilesN * tilesN) return;
  const int tm = tid / tilesN;
  const int tn = tid - tm * tilesN;
  const int row = (tm << 4) + l16;
  const int jrow = (tn << 4) + l16;

  const float* ap = S + (size_t)row * lds + (hl << 1);
  const float* bp = S + (size_t)jrow * lds + (hl << 1);

  v8f acc = {};
#pragma unroll 4
  for (int k0 = 0; k0 < Kd; k0 += 4) {
    v2f a = *(const v2f*)ap;
    v2f b = *(const v2f*)bp;
    acc = wmma_f32(a, b, acc);
    ap += 4;
    bp += 4;
  }
  float part = 0.0f;
#pragma unroll
  for (int r = 0; r < 8; ++r) {
    const int rr = (tm << 4) + r + (hl << 3);
    const float d = adj[(size_t)rr * M + jrow] - acc[r];
    part += d * d;
  }
  for (int off = 16; off > 0; off >>= 1) part += __shfl_down(part, off, 32);
  if (lane == 0) atomicAdd(acc_out, part);
}

// ---------------------------------------------------------------------------
// invdeg[i] = 1 / max(rowsum(adj[i,:]) + selfadd, 1)     (one wave per row)
// ---------------------------------------------------------------------------
__global__ void k_invdeg(const float* __restrict__ adj, float* __restrict__ invdeg,
                         int M, int N, float selfadd) {
  const int wid = (blockIdx.x * blockDim.x + threadIdx.x) >> 5;
  const int lane = threadIdx.x & 31;
  if (wid >= M) return;
  const float* row = adj + (size_t)wid * N;
  float s = 0.0f;
  for (int j = lane; j < N; j += 32) s += row[j];
  for (int off = 16; off > 0; off >>= 1) s += __shfl_down(s, off, 32);
  if (lane == 0) invdeg[wid] = 1.0f / fmaxf(s + selfadd, 1.0f);
}

// ---------------------------------------------------------------------------
// Row softmax in place + entropy accumulation. One wave per row.
// ---------------------------------------------------------------------------
__global__ void k_softmax_ent(float* __restrict__ S, int M, int N,
                              float* __restrict__ ent_acc) {
  const int wid = (blockIdx.x * blockDim.x + threadIdx.x) >> 5;
  const int lane = threadIdx.x & 31;
  if (wid >= M) return;
  float* row = S + (size_t)wid * N;
  float mx = -3.4e38f;
  for (int j = lane; j < N; j += 32) mx = fmaxf(mx, row[j]);
  for (int off = 16; off > 0; off >>= 1) mx = fmaxf(mx, __shfl_xor(mx, off, 32));
  float sum = 0.0f;
  for (int j = lane; j < N; j += 32) sum += expf(row[j] - mx);
  for (int off = 16; off > 0; off >>= 1) sum += __shfl_xor(sum, off, 32);
  const float inv = 1.0f / sum;
  float ent = 0.0f;
  for (int j = lane; j < N; j += 32) {
    const float p = expf(row[j] - mx) * inv;
    row[j] = p;
    ent -= p * logf(p + 1e-15f);
  }
  for (int off = 16; off > 0; off >>= 1) ent += __shfl_down(ent, off, 32);
  if (lane == 0) atomicAdd(ent_acc, ent);
}

// g[c] = mean over M rows of H[:, c]
__global__ void k_mean(const float* __restrict__ H, int ld, int M, int C,
                       float* __restrict__ g) {
  const int c = blockIdx.x * blockDim.x + threadIdx.x;
  if (c >= C) return;
  float s = 0.0f;
  for (int m = 0; m < M; ++m) s += H[(size_t)m * ld + c];
  g[c] = s / (float)M;
}

// ---------------------------------------------------------------------------
// Head: struct linear -> joint feature synthesis -> output packing.
// scal = [link1, ent1, link2, ent2]. out: [20 outputs][loss][ent][20*128 embeds]
// ---------------------------------------------------------------------------
__global__ __launch_bounds__(128) void k_head(
    const float* __restrict__ g, const float* __restrict__ sw,
    const float* __restrict__ sb, const float* __restrict__ jw_noise,
    const float* __restrict__ out_w, const float* __restrict__ out_b,
    const float* __restrict__ scal, float* __restrict__ out) {
  __shared__ float base[128];
  __shared__ float fe_s[20][128];
  const int d = threadIdx.x;  // 128 threads, D = 128
  float acc = sb[d];
  for (int c = 0; c < 192; ++c) acc += g[c] * sw[c * 128 + d];
  base[d] = fmaxf(acc, 0.0f);
  __syncthreads();
  const float b0 = base[0], b1 = base[1];
  for (int j = 0; j < 20; ++j) {
    const float fi = (float)j;
    const float ang = (fi + 1.0f) * 0.5f;
    const float cs = cosf(ang), sn = sinf(ang);
    const float scale = 1.0f + fi * 0.1f;
    float t = base[d];
    if (d == 0) t = cs * b0 - sn * b1;
    else if (d == 1) t = sn * b0 + cs * b1;
    t *= scale;
    const int m = d & 3;
    const float off = (m == 0) ? fi * 0.2f
                    : (m == 1) ? sinf(fi * 0.5f)
                    : (m == 2) ? cosf(fi * 0.7f)
                               : fi * fi * 0.05f;
    const float freq = sinf((fi + 1.0f) * ((float)d + 1.0f) * 0.01f);
    float fe = tanhf(t + off * 0.3f + freq * 0.2f);
    const float jw = (d == (j % 128)) ? 1.0f : jw_noise[j * 128 + d] * 0.1f;
    fe *= (1.0f + jw);
    fe_s[j][d] = fe;
    out[22 + j * 128 + d] = fe;  // joint_embeds
  }
  __syncthreads();
  if (d < 20) {
    float s = 0.0f;
    for (int c = 0; c < 128; ++c) s += fe_s[d][c] * out_w[c];
    out[d] = s + out_b[0];
  }
  if (d == 0) {
    out[20] = sqrtf(scal[0]) / (4096.0f * 4096.0f) +
              sqrtf(scal[2]) / (1024.0f * 1024.0f);
    out[21] = scal[1] / 4096.0f + scal[3] / 1024.0f;
  }
}

// ---------------------------------------------------------------------------
// Host-side launch helpers
// ---------------------------------------------------------------------------
static inline void gemm(hipStream_t st, const float* A, int lda, int transA,
                        const float* B, int ldb, const float* bias, float* D,
                        int ldd, int M, int N, int K, int addB, int relu) {
  const int tiles = (M / 16) * (N / 16);
  const int blocks = (tiles + 7) / 8;  // 8 waves per block
  if (transA) {
    k_gemm_t<true, false, false, false>
        <<<blocks, 256, 0, st>>>(A, lda, B, ldb, nullptr, D, ldd, M, N, K);
  } else if (addB) {
    k_gemm_t<false, true, false, false>
        <<<blocks, 256, 0, st>>>(A, lda, B, ldb, nullptr, D, ldd, M, N, K);
  } else if (bias) {
    (void)relu;  // bias is always paired with relu in this pipeline
    k_gemm_t<false, false, true, true>
        <<<blocks, 256, 0, st>>>(A, lda, B, ldb, bias, D, ldd, M, N, K);
  } else {
    k_gemm_t<false, false, false, false>
        <<<blocks, 256, 0, st>>>(A, lda, B, ldb, nullptr, D, ldd, M, N, K);
  }
}

static inline void sage(hipStream_t st, const float* Agg, int ldAgg,
                        const float* X, int ldX, const float* invdeg,
                        const float* Wrel, const float* Wroot, const float* b,
                        float* Out, int ldout, int M, int N, int K) {
  const int tiles = (M / 16) * (N / 16);
  const int blocks = (tiles + 7) / 8;
  k_sage<<<blocks, 256, 0, st>>>(Agg, ldAgg, X, ldX, invdeg, Wrel, Wroot, b,
                                 Out, ldout, M, N, K);
}

extern "C" void kernel_launch(void* const* d_in, const int* in_sizes, int n_in,
                              void* d_out, int out_size, void* d_ws,
                              size_t ws_size, hipStream_t stream) {
  (void)in_sizes; (void)n_in; (void)out_size; (void)ws_size;
#define IN(i) ((const float*)d_in[(i)])
  // Input flattening assumed in setup_inputs() insertion order (recursive):
  // 0:x 1:adj
  // gnn1_pool: 2..4 c1(wr,wo,b) 5..7 c2 8..10 c3 11:lin_w 12:lin_b
  // gnn1_embed: 13..15 c1 16..18 c2 19..21 c3
  // gnn2_pool: 22..24 c1 25..27 c2 28..30 c3 31:lin_w 32:lin_b
  // gnn2_embed: 33..35 36..38 39..41
  // gnn3_embed: 42..44 45..47 48..50
  // 51:struct_w 52:struct_b 53:out_w 54:out_b 55:jw_noise 56:num_joints
  const float* x = IN(0);
  const float* adj = IN(1);
  float* out = (float*)d_out;

  const int N0 = 4096, N1 = 1024, N2 = 256;

  // Workspace layout (floats)
  float* ws = (float*)d_ws;
  size_t o = 0;
  float* aggA  = ws + o; o += (size_t)N0 * 64;    // shared c1 aggregation
  float* aggB  = ws + o; o += (size_t)N0 * 64;    // scratch aggregation (>=1024*192)
  float* catP1 = ws + o; o += (size_t)N0 * 1152;  // [x1p|x2p|x3p]
  float* catE1 = ws + o; o += (size_t)N0 * 192;   // h1 = [x1e|x2e|x3e]
  float* sbuf  = ws + o; o += (size_t)N0 * 1024;  // s (stage1) / s2 (stage2)
  float* tmpB  = ws + o; o += (size_t)N1 * 4096;  // s^T @ adj scratch
  float* adj2  = ws + o; o += (size_t)N1 * N1;
  float* adj3  = ws + o; o += (size_t)N2 * N2;
  float* h1p   = ws + o; o += (size_t)N1 * 192;   // pooled features stage 1
  float* h2    = ws + o; o += (size_t)N2 * 192;   // pooled features stage 2
  float* catP2 = ws + o; o += (size_t)N1 * 384;
  float* catE2 = ws + o; o += (size_t)N1 * 192;
  float* catE3 = ws + o; o += (size_t)N2 * 192;
  float* ideg1 = ws + o; o += N0;
  float* ideg2 = ws + o; o += N1;
  float* ideg3 = ws + o; o += N2;
  float* gvec  = ws + o; o += 192;
  float* scal  = ws + o; o += 4;  // [link1, ent1, link2, ent2]

  hipMemsetAsync(scal, 0, 4 * sizeof(float), stream);

  // ============================ Stage 1 (N=4096) ============================
  // adj1 = adj + I  =>  agg = adj@X + X, deg = rowsum(adj)+1
  k_invdeg<<<(N0 + 3) / 4, 128, 0, stream>>>(adj, ideg1, N0, N0, 1.0f);
  gemm(stream, adj, N0, 0, x, 64, nullptr, aggA, 64, N0, 64, N0, 1, 0);
  // pool c1 / embed c1 share aggA
  sage(stream, aggA, 64, x, 64, ideg1, IN(2), IN(3), IN(4), catP1, 1152, N0, 64, 64);
  sage(stream, aggA, 64, x, 64, ideg1, IN(13), IN(14), IN(15), catE1, 192, N0, 64, 64);
  // pool c2, c3
  gemm(stream, adj, N0, 0, catP1, 1152, nullptr, aggB, 64, N0, 64, N0, 1, 0);
  sage(stream, aggB, 64, catP1, 1152, ideg1, IN(5), IN(6), IN(7), catP1 + 64, 1152, N0, 64, 64);
  gemm(stream, adj, N0, 0, catP1 + 64, 1152, nullptr, aggB, 64, N0, 64, N0, 1, 0);
  sage(stream, aggB, 64, catP1 + 64, 1152, ideg1, IN(8), IN(9), IN(10), catP1 + 128, 1152, N0, 1024, 64);
  // embed c2, c3
  gemm(stream, adj, N0, 0, catE1, 192, nullptr, aggB, 64, N0, 64, N0, 1, 0);
  sage(stream, aggB, 64, catE1, 192, ideg1, IN(16), IN(17), IN(18), catE1 + 64, 192, N0, 64, 64);
  gemm(stream, adj, N0, 0, catE1 + 64, 192, nullptr, aggB, 64, N0, 64, N0, 1, 0);
  sage(stream, aggB, 64, catE1 + 64, 192, ideg1, IN(19), IN(20), IN(21), catE1 + 128, 192, N0, 64, 64);
  // pool lin + softmax (s), entropy e1
  gemm(stream, catP1, 1152, 0, IN(11), 1024, IN(12), sbuf, 1024, N0, 1024, 1152, 0, 1);
  k_softmax_ent<<<(N0 + 3) / 4, 128, 0, stream>>>(sbuf, N0, 1024, scal + 1);
  // diff_pool 1
  gemm(stream, sbuf, 1024, 1, catE1, 192, nullptr, h1p, 192, N1, 192, N0, 0, 0);   // s^T @ h1
  gemm(stream, sbuf, 1024, 1, adj, N0, nullptr, tmpB, N0, N1, N0, N0, 0, 0);       // s^T @ adj
  gemm(stream, tmpB, N0, 0, sbuf, 1024, nullptr, adj2, N1, N1, N1, N0, 0, 0);      // (s^T adj) s
  {
    const int tiles = (N0 / 16) * (N0 / 16);
    k_linkloss<<<(tiles + 7) / 8, 256, 0, stream>>>(sbuf, 1024, adj, N0, 1024, scal + 0);
  }

  // ============================ Stage 2 (N=1024) ============================
  k_invdeg<<<(N1 + 3) / 4, 128, 0, stream>>>(adj2, ideg2, N1, N1, 0.0f);
  gemm(stream, adj2, N1, 0, h1p, 192, nullptr, aggA, 192, N1, 192, N1, 0, 0);
  sage(stream, aggA, 192, h1p, 192, ideg2, IN(22), IN(23), IN(24), catP2, 384, N1, 64, 192);
  sage(stream, aggA, 192, h1p, 192, ideg2, IN(33), IN(34), IN(35), catE2, 192, N1, 64, 192);
  gemm(stream, adj2, N1, 0, catP2, 384, nullptr, aggB, 64, N1, 64, N1, 0, 0);
  sage(stream, aggB, 64, catP2, 384, ideg2, IN(25), IN(26), IN(27), catP2 + 64, 384, N1, 64, 64);
  gemm(stream, adj2, N1, 0, catP2 + 64, 384, nullptr, aggB, 64, N1, 64, N1, 0, 0);
  sage(stream, aggB, 64, catP2 + 64, 384, ideg2, IN(28), IN(29), IN(30), catP2 + 128, 384, N1, 256, 64);
  gemm(stream, adj2, N1, 0, catE2, 192, nullptr, aggB, 64, N1, 64, N1, 0, 0);
  sage(stream, aggB, 64, catE2, 192, ideg2, IN(36), IN(37), IN(38), catE2 + 64, 192, N1, 64, 64);
  gemm(stream, adj2, N1, 0, catE2 + 64, 192, nullptr, aggB, 64, N1, 64, N1, 0, 0);
  sage(stream, aggB, 64, catE2 + 64, 192, ideg2, IN(39), IN(40), IN(41), catE2 + 128, 192, N1, 64, 64);
  gemm(stream, catP2, 384, 0, IN(31), 256, IN(32), sbuf, 256, N1, 256, 384, 0, 1);
  k_softmax_ent<<<(N1 + 3) / 4, 128, 0, stream>>>(sbuf, N1, 256, scal + 3);
  // diff_pool 2
  gemm(stream, sbuf, 256, 1, catE2, 192, nullptr, h2, 192, N2, 192, N1, 0, 0);
  gemm(stream, sbuf, 256, 1, adj2, N1, nullptr, tmpB, N1, N2, N1, N1, 0, 0);
  gemm(stream, tmpB, N1, 0, sbuf, 256, nullptr, adj3, N2, N2, N2, N1, 0, 0);
  {
    const int tiles = (N1 / 16) * (N1 / 16);
    k_linkloss<<<(tiles + 7) / 8, 256, 0, stream>>>(sbuf, 256, adj2, N1, 256, scal + 2);
  }

  // ============================ Stage 3 (N=256) =============================
  k_invdeg<<<(N2 + 3) / 4, 128, 0, stream>>>(adj3, ideg3, N2, N2, 0.0f);
  gemm(stream, adj3, N2, 0, h2, 192, nullptr, aggA, 192, N2, 192, N2, 0, 0);
  sage(stream, aggA, 192, h2, 192, ideg3, IN(42), IN(43), IN(44), catE3, 192, N2, 64, 192);
  gemm(stream, adj3, N2, 0, catE3, 192, nullptr, aggB, 64, N2, 64, N2, 0, 0);
  sage(stream, aggB, 64, catE3, 192, ideg3, IN(45), IN(46), IN(47), catE3 + 64, 192, N2, 64, 64);
  gemm(stream, adj3, N2, 0, catE3 + 64, 192, nullptr, aggB, 64, N2, 64, N2, 0, 0);
  sage(stream, aggB, 64, catE3 + 64, 192, ideg3, IN(48), IN(49), IN(50), catE3 + 128, 192, N2, 64, 64);

  // ============================ Head ========================================
  k_mean<<<3, 64, 0, stream>>>(catE3, 192, N2, 192, gvec);
  k_head<<<1, 128, 0, stream>>>(gvec, IN(51), IN(52), IN(55), IN(53), IN(54),
                                scal, out);
#undef IN
}